// ParameterTypeEmbedding_5446018531762
// MI455X (gfx1250) — compile-verified
//
#include <hip/hip_runtime.h>
#include <hip/hip_bf16.h>
#include <math.h>

// ---------------------------------------------------------------------------
// ParameterTypeEmbedding on MI455X (gfx1250).
// Key insight: LayerNorm(1) zeroes the data path -> every (b,n) row of the
// output is the same 896-float vector. Compute it ONCE (WMMA f32 16x16x4),
// then the problem is a pure 1.43 GB broadcast store (~62us at 23.3 TB/s).
// ---------------------------------------------------------------------------

typedef __attribute__((ext_vector_type(2))) float v2f;
typedef __attribute__((ext_vector_type(4))) float v4f;   // clang ext-vector: OK for NT builtins
typedef __attribute__((ext_vector_type(8))) float v8f;

#define F_DIM  14
#define H1D    16
#define H2D    32
#define H3D    16
#define E_DIM  64
#define ROW_F  (F_DIM * E_DIM)   // 896 floats per output row
#define ROW_F4 (ROW_F / 4)       // 224 float4 per output row

__device__ __forceinline__ float gelu_exact(float x) {
    return 0.5f * x * (1.0f + erff(x * 0.70710678118654752440f));
}

// One 16-wide output tile of  row0(A)=vec[K]  times  W[K x ldw] (cols ncol0..+15)
// using a chain of V_WMMA_F32_16X16X4_F32.
// A 16x4 f32 layout: lane0 v0/v1 = (M=0,K=0/1); lane16 v0/v1 = (M=0,K=2/3);
// rows M=1..15 are zero, so D rows 1..15 stay zero.
// B 4x16 f32 layout (rows striped across lane halves, mirroring A/CD pattern):
// v0: lanes0-15 K=0, lanes16-31 K=2 ; v1: lanes0-15 K=1, lanes16-31 K=3.
// Result row 0 lands in D element 0, lanes 0..15 (N = lane).
template <int K>
__device__ __forceinline__ v8f wmma_vecmat_tile(const float* __restrict__ vec_lds,
                                                const float* __restrict__ W,
                                                int ldw, int ncol0, int lane) {
    v8f acc;
    acc[0] = 0.f; acc[1] = 0.f; acc[2] = 0.f; acc[3] = 0.f;
    acc[4] = 0.f; acc[5] = 0.f; acc[6] = 0.f; acc[7] = 0.f;
    const bool hi  = lane >= 16;
    const int  l16 = lane & 15;
#pragma unroll
    for (int kb = 0; kb < K; kb += 4) {
        v2f a;
        a.x = (lane == 0) ? vec_lds[kb + 0] : ((lane == 16) ? vec_lds[kb + 2] : 0.0f);
        a.y = (lane == 0) ? vec_lds[kb + 1] : ((lane == 16) ? vec_lds[kb + 3] : 0.0f);
        const int k0 = kb + (hi ? 2 : 0);
        v2f b;
        b.x = W[(k0 + 0) * ldw + ncol0 + l16];
        b.y = W[(k0 + 1) * ldw + ncol0 + l16];
        acc = __builtin_amdgcn_wmma_f32_16x16x4_f32(false, a, false, b,
                                                    (short)0, acc, false, false);
    }
    return acc;
}

// ---------------------------------------------------------------------------
// Kernel A: build the 896-float table. 14 blocks (one feature each) x 1 wave.
// Block size 32 == one wave32 -> EXEC all-1s for WMMA as required.
// ---------------------------------------------------------------------------
__global__ __launch_bounds__(32)
void ptemb_table_kernel(const float* __restrict__ beta,
                        const float* __restrict__ W1, const float* __restrict__ b1,
                        const float* __restrict__ W2, const float* __restrict__ b2,
                        const float* __restrict__ W3, const float* __restrict__ b3,
                        const float* __restrict__ W4, const float* __restrict__ b4,
                        const float* __restrict__ emb,
                        float* __restrict__ table) {
    const int f    = blockIdx.x;    // feature 0..13
    const int lane = threadIdx.x;   // 0..31

    __shared__ float h1s[H1D];
    __shared__ float h2s[H2D];
    __shared__ float h3s[H3D];

    // Layer 1: LayerNorm(1) output is just beta[f]; scalar -> H1.
    const float bf = beta[f];
    if (lane < H1D) {
        const float t = bf * W1[f * H1D + lane] + b1[f * H1D + lane];
        h1s[lane] = gelu_exact(t);
    }
    __syncthreads();

    // Layer 2: [1xH1] @ W2[f] (16x32) -> h2[32].  2 tiles, K=16.
#pragma unroll
    for (int t = 0; t < 2; ++t) {
        v8f acc = wmma_vecmat_tile<H1D>(h1s, W2 + f * H1D * H2D, H2D, t * 16, lane);
        if (lane < 16) {
            const int n = t * 16 + lane;
            h2s[n] = gelu_exact(acc[0] + b2[f * H2D + n]);
        }
        __syncthreads();
    }

    // Layer 3: [1xH2] @ W3[f] (32x16) -> h3[16].  1 tile, K=32.
    {
        v8f acc = wmma_vecmat_tile<H2D>(h2s, W3 + f * H2D * H3D, H3D, 0, lane);
        if (lane < 16) {
            h3s[lane] = gelu_exact(acc[0] + b3[f * H3D + lane]);
        }
        __syncthreads();
    }

    // Layer 4: [1xH3] @ W4[f] (16x64) + b4 + emb -> table[f*64 .. +63].  4 tiles.
#pragma unroll
    for (int t = 0; t < 4; ++t) {
        v8f acc = wmma_vecmat_tile<H3D>(h3s, W4 + f * H3D * E_DIM, E_DIM, t * 16, lane);
        if (lane < 16) {
            const int e = t * 16 + lane;
            table[f * E_DIM + e] = acc[0] + b4[f * E_DIM + e] + emb[f * E_DIM + e];
        }
    }
}

// ---------------------------------------------------------------------------
// Kernel B: broadcast the 896-float row to all B*N rows with NT B128 stores.
// Each thread owns one float4 column (kept in a register); each block fills
// ROWS_PER_BLOCK contiguous rows -> 224 KB contiguous per block, fully
// coalesced 3584 B per row per iteration. NT hint: 1.43 GB >> 192 MB L2.
// ---------------------------------------------------------------------------
#define BCAST_THREADS  224
#define ROWS_PER_BLOCK 64

__global__ __launch_bounds__(BCAST_THREADS)
void ptemb_bcast_kernel(const v4f* __restrict__ table4,
                        v4f* __restrict__ out4, int n_rows) {
    const int c = threadIdx.x;              // float4 column 0..223
    const v4f v = table4[c];                // tiny read, hits L2

    long r0 = (long)blockIdx.x * ROWS_PER_BLOCK;
    long r1 = r0 + ROWS_PER_BLOCK;
    if (r1 > (long)n_rows) r1 = (long)n_rows;

    v4f* p = out4 + r0 * (long)ROW_F4 + c;
    for (long r = r0; r < r1; ++r, p += ROW_F4) {
        __builtin_nontemporal_store(v, p);
    }
}

// ---------------------------------------------------------------------------
// Launch. Input order: x, gamma, beta, W1, b1, W2, b2, W3, b3, W4, b4, emb.
// x and gamma are mathematically unused (LayerNorm(1) degeneracy).
// ---------------------------------------------------------------------------
extern "C" void kernel_launch(void* const* d_in, const int* in_sizes, int n_in,
                              void* d_out, int out_size, void* d_ws, size_t ws_size,
                              hipStream_t stream) {
    (void)in_sizes; (void)n_in; (void)ws_size;
    const float* beta = (const float*)d_in[2];
    const float* W1   = (const float*)d_in[3];
    const float* b1   = (const float*)d_in[4];
    const float* W2   = (const float*)d_in[5];
    const float* b2   = (const float*)d_in[6];
    const float* W3   = (const float*)d_in[7];
    const float* b3   = (const float*)d_in[8];
    const float* W4   = (const float*)d_in[9];
    const float* b4   = (const float*)d_in[10];
    const float* emb  = (const float*)d_in[11];

    float* table = (float*)d_ws;  // 896 floats of scratch

    ptemb_table_kernel<<<F_DIM, 32, 0, stream>>>(beta, W1, b1, W2, b2, W3, b3,
                                                 W4, b4, emb, table);

    const int n_rows = out_size / ROW_F;  // 400000
    const int grid   = (n_rows + ROWS_PER_BLOCK - 1) / ROWS_PER_BLOCK;
    ptemb_bcast_kernel<<<grid, BCAST_THREADS, 0, stream>>>(
        (const v4f*)d_ws, (v4f*)d_out, n_rows);
}